// AdaptiveBlockSparseAttnTrain_53111565582693
// MI455X (gfx1250) — compile-verified
//
#include <hip/hip_runtime.h>

#define H_      12
#define S_      4096
#define D_      128
#define DP_     132      // padded LDS row stride (132 mod 64 = 4 -> conflict-free strided reads)
#define BLOCK_  128
#define NB_     32
#define NKEEP_  32
#define MAXRET_ 5

typedef __attribute__((ext_vector_type(2))) float v2f;
typedef __attribute__((ext_vector_type(8))) float v8f;
typedef __attribute__((__vector_size__(4*sizeof(int)))) int v4i;

static __device__ __forceinline__ v8f vzero8() {
  v8f z = {0.f,0.f,0.f,0.f,0.f,0.f,0.f,0.f};
  return z;
}

// ---- CDNA5 async global->LDS copy path (GLOBAL_LOAD_ASYNC_TO_LDS_B128) ----
#if defined(__has_builtin)
#if __has_builtin(__builtin_amdgcn_global_load_async_to_lds_b128)
#define ASYNC_LDS 1
#endif
#endif
#ifndef ASYNC_LDS
#define ASYNC_LDS 0
#endif

#if ASYNC_LDS
static __device__ __forceinline__ void async_cp16(const float* g, float* l) {
  __builtin_amdgcn_global_load_async_to_lds_b128(
      (__attribute__((address_space(1))) v4i*)(float*)g,   // global src (v4i*, AS1)
      (__attribute__((address_space(3))) v4i*)l,           // LDS dst    (v4i*, AS3)
      0, 0);
}
static __device__ __forceinline__ void wait_async0() {
#if __has_builtin(__builtin_amdgcn_s_wait_asynccnt)
  __builtin_amdgcn_s_wait_asynccnt(0);
#else
  asm volatile("s_wait_asynccnt 0" ::: "memory");
#endif
}
#endif

// ---------------------------------------------------------------------------
// Stage 1: sampled-token attention pooling -> retained key-block list per
// (head, q-block). One workgroup per (h, qb); 256 threads.
// ---------------------------------------------------------------------------
__global__ __launch_bounds__(256) void absa_mask_kernel(
    const float* __restrict__ q, const float* __restrict__ k,
    const int* __restrict__ idx_q, const int* __restrict__ idx_k,
    int* __restrict__ ret_cnt, int* __restrict__ ret_idx)
{
  const int h  = blockIdx.x / NB_;
  const int qb = blockIdx.x % NB_;
  const int t  = threadIdx.x;

  __shared__ float sq[NKEEP_][D_];           // 16 KB
  __shared__ float sc[NKEEP_][NB_*NKEEP_];   // 128 KB (needs CDNA5's 320 KB LDS)
  __shared__ float rowmax[NKEEP_], rowsum[NKEEP_], pool[NB_];

  // gather the 32 sampled q rows of this query block
  for (int e = t; e < NKEEP_*D_; e += 256) {
    int qi = e >> 7, d = e & (D_-1);
    int tok = qb*BLOCK_ + idx_q[h*NKEEP_ + qi];
    sq[qi][d] = q[((size_t)h*S_ + tok)*D_ + d];
  }
  __syncthreads();

  const float scale = 0.088388347648318447f;  // 1/sqrt(128)

  // scores: 32 sampled q rows vs all 1024 sampled k tokens
  for (int c = t; c < NB_*NKEEP_; c += 256) {
    int kb = c >> 5, j = c & 31;
    int tok = kb*BLOCK_ + idx_k[h*NKEEP_ + j];
    const float* kp = &k[((size_t)h*S_ + tok)*D_];
    float acc[NKEEP_];
#pragma unroll
    for (int qi = 0; qi < NKEEP_; ++qi) acc[qi] = 0.f;
    for (int d = 0; d < D_; ++d) {
      float kv = kp[d];
#pragma unroll
      for (int qi = 0; qi < NKEEP_; ++qi) acc[qi] = fmaf(sq[qi][d], kv, acc[qi]);
    }
#pragma unroll
    for (int qi = 0; qi < NKEEP_; ++qi) sc[qi][c] = acc[qi]*scale;
  }
  __syncthreads();

  // per-row softmax stats over all 1024 sampled keys
  if (t < NKEEP_) {
    float mx = -INFINITY;
    for (int c = 0; c < NB_*NKEEP_; ++c) mx = fmaxf(mx, sc[t][c]);
    float s = 0.f;
    for (int c = 0; c < NB_*NKEEP_; ++c) s += __expf(sc[t][c]-mx);
    rowmax[t] = mx; rowsum[t] = s;
  }
  __syncthreads();

  // pool attention mass per key block
  if (t < NB_) {
    float p = 0.f;
    for (int qi = 0; qi < NKEEP_; ++qi) {
      float mx = rowmax[qi], inv = 1.f/rowsum[qi];
      float ps = 0.f;
      for (int j = 0; j < NKEEP_; ++j) ps += __expf(sc[qi][t*NKEEP_+j]-mx);
      p = fmaf(ps, inv, p);
    }
    pool[t] = p;
  }
  __syncthreads();

  // descending selection + cumsum threshold (matches reference argsort/argmax)
  if (t == 0) {
    float pv[NB_]; bool used[NB_];
    float total = 0.f;
    for (int i = 0; i < NB_; ++i) { pv[i] = pool[i]; used[i] = false; total += pv[i]; }
    const float thr = 0.95f * total;
    int sel[NB_];
    float cum = 0.f;
    int kidx = NB_;
    for (int i = 0; i < NB_; ++i) {
      int best = -1; float bv = -INFINITY;
      for (int j = 0; j < NB_; ++j)
        if (!used[j] && pv[j] > bv) { bv = pv[j]; best = j; }
      used[best] = true; sel[i] = best; cum += bv;
      if (kidx == NB_ && cum >= thr) kidx = i;
    }
    int cnt = kidx; if (cnt < 1) cnt = 1; if (cnt > MAXRET_) cnt = MAXRET_;
    ret_cnt[h*NB_+qb] = cnt;
    for (int i = 0; i < 8; ++i)
      ret_idx[(h*NB_+qb)*8 + i] = sel[(i < cnt) ? i : 0];
  }
}

// ---------------------------------------------------------------------------
// Stage 2: block-sparse flash attention over retained key blocks.
// One workgroup per (h, qb); 8 wave32s; each wave owns a 16-row q strip.
// WMMA f32 16x16x4 for both Q*K^T and P*V (full f32 fidelity).
// Tiles staged in LDS via GLOBAL_LOAD_ASYNC_TO_LDS_B128 when available.
// ---------------------------------------------------------------------------
__global__ __launch_bounds__(256) void absa_attn_kernel(
    const float* __restrict__ q, const float* __restrict__ k, const float* __restrict__ v,
    const int* __restrict__ ret_cnt, const int* __restrict__ ret_idx,
    float* __restrict__ out)
{
  const int h     = blockIdx.x / NB_;
  const int qb    = blockIdx.x % NB_;
  const int t     = threadIdx.x;
  const int wave  = t >> 5;
  const int lane  = t & 31;
  const int m16   = lane & 15;
  const int half  = lane >> 4;      // 0 or 1 (half-wave)
  const int khalf = half << 1;      // K offset 0 or 2 for A/B fragments
  const int rows0 = wave << 4;      // q-strip base row within the 128-row tile

  __shared__ float Qs[BLOCK_][DP_];   // ~66 KB
  __shared__ float KPs[BLOCK_][DP_];  // K, then reused for P  (~66 KB)
  __shared__ float Vs[BLOCK_][DP_];   // ~66 KB   (total ~198 KB of 320 KB)

  // load Q tile (async global->LDS if available, else float4 staging)
  {
    const float* qp = &q[((size_t)h*S_ + (size_t)qb*BLOCK_)*D_];
    for (int idx = t; idx < BLOCK_*(D_/4); idx += 256) {
      int r = idx >> 5, c = (idx & 31) << 2;
#if ASYNC_LDS
      async_cp16(&qp[r*D_ + c], &Qs[r][c]);
#else
      *(float4*)&Qs[r][c] = *(const float4*)&qp[r*D_ + c];
#endif
    }
  }

  v8f   acc_o[8];
  float rm[8], rl[8];
#pragma unroll
  for (int i = 0; i < 8; ++i) { acc_o[i] = vzero8(); rm[i] = -INFINITY; rl[i] = 0.f; }

  const int   cnt   = ret_cnt[h*NB_+qb];
  const float scale = 0.088388347648318447f;

  for (int it = 0; it < cnt; ++it) {
    const int kb = ret_idx[(h*NB_+qb)*8 + it];
    __syncthreads();   // prior iteration's P/V reads complete before reload
    {
      const float* kp = &k[((size_t)h*S_ + (size_t)kb*BLOCK_)*D_];
      const float* vp = &v[((size_t)h*S_ + (size_t)kb*BLOCK_)*D_];
      for (int idx = t; idx < BLOCK_*(D_/4); idx += 256) {
        int r = idx >> 5, c = (idx & 31) << 2;
#if ASYNC_LDS
        async_cp16(&kp[r*D_ + c], &KPs[r][c]);
        async_cp16(&vp[r*D_ + c], &Vs[r][c]);
#else
        *(float4*)&KPs[r][c] = *(const float4*)&kp[r*D_ + c];
        *(float4*)&Vs[r][c]  = *(const float4*)&vp[r*D_ + c];
#endif
      }
    }
    if (it + 1 < cnt) {   // prefetch next retained block (global_prefetch_b8)
      const int kb2 = ret_idx[(h*NB_+qb)*8 + it + 1];
      const float* kn = &k[((size_t)h*S_ + (size_t)kb2*BLOCK_)*D_];
      const float* vn = &v[((size_t)h*S_ + (size_t)kb2*BLOCK_)*D_];
      __builtin_prefetch(kn + (size_t)t*64, 0, 3);
      __builtin_prefetch(vn + (size_t)t*64, 0, 3);
    }
#if ASYNC_LDS
    wait_async0();       // ASYNCcnt==0: this wave's LDS writes landed
#endif
    __syncthreads();     // all waves' tile data visible

    // ---- S = Q_strip(16x128) * K^T(128x128) via WMMA f32 16x16x4 ----
    v8f accs[8];
#pragma unroll
    for (int nt = 0; nt < 8; ++nt) accs[nt] = vzero8();
    for (int k0 = 0; k0 < D_; k0 += 4) {
      // A fragment: lane holds A[m16][k0+khalf .. +1]   (ISA 16x4 f32 layout)
      v2f a; a.x = Qs[rows0+m16][k0+khalf]; a.y = Qs[rows0+m16][k0+khalf+1];
#pragma unroll
      for (int nt = 0; nt < 8; ++nt) {
        // B fragment: B[k][n] = K[nt*16+n][k]
        v2f b; b.x = KPs[nt*16+m16][k0+khalf]; b.y = KPs[nt*16+m16][k0+khalf+1];
        accs[nt] = __builtin_amdgcn_wmma_f32_16x16x4_f32(
            false, a, false, b, (short)0, accs[nt], false, false);
      }
    }
#pragma unroll
    for (int nt = 0; nt < 8; ++nt)
#pragma unroll
      for (int i = 0; i < 8; ++i) accs[nt][i] *= scale;

    // ---- online softmax: row r of C-tile lives in lanes of one half ----
#pragma unroll
    for (int r = 0; r < 8; ++r) {
      float mr = -INFINITY;
#pragma unroll
      for (int nt = 0; nt < 8; ++nt) mr = fmaxf(mr, accs[nt][r]);
#pragma unroll
      for (int off = 8; off >= 1; off >>= 1) mr = fmaxf(mr, __shfl_xor(mr, off, 32));
      float mnew = fmaxf(rm[r], mr);
      float corr = __expf(rm[r] - mnew);
      rm[r] = mnew;
      float sr = 0.f;
#pragma unroll
      for (int nt = 0; nt < 8; ++nt) {
        float e = __expf(accs[nt][r] - mnew);
        accs[nt][r] = e;
        sr += e;
      }
#pragma unroll
      for (int off = 8; off >= 1; off >>= 1) sr += __shfl_xor(sr, off, 32);
      rl[r] = rl[r]*corr + sr;
#pragma unroll
      for (int dt = 0; dt < 8; ++dt) acc_o[dt][r] *= corr;
    }

    __syncthreads();   // everyone is done reading K from KPs
    // spill P into KPs (each wave writes only its own 16-row slice)
#pragma unroll
    for (int nt = 0; nt < 8; ++nt)
#pragma unroll
      for (int r = 0; r < 8; ++r)
        KPs[rows0 + (half<<3) + r][nt*16 + m16] = accs[nt][r];
    __syncthreads();

    // ---- acc_o += P(16x128) * V(128x128) via WMMA f32 16x16x4 ----
    for (int kk = 0; kk < BLOCK_; kk += 4) {
      v2f a; a.x = KPs[rows0+m16][kk+khalf]; a.y = KPs[rows0+m16][kk+khalf+1];
#pragma unroll
      for (int dt = 0; dt < 8; ++dt) {
        v2f b; b.x = Vs[kk+khalf][dt*16+m16]; b.y = Vs[kk+khalf+1][dt*16+m16];
        acc_o[dt] = __builtin_amdgcn_wmma_f32_16x16x4_f32(
            false, a, false, b, (short)0, acc_o[dt], false, false);
      }
    }
  }

  // epilogue: normalize by row sum and store
  float* op = &out[((size_t)h*S_ + (size_t)qb*BLOCK_ + rows0)*D_];
#pragma unroll
  for (int r = 0; r < 8; ++r) {
    float inv = 1.f / rl[r];
    int row = (half<<3) + r;
#pragma unroll
    for (int dt = 0; dt < 8; ++dt)
      op[row*D_ + dt*16 + m16] = acc_o[dt][r] * inv;
  }
}

extern "C" void kernel_launch(void* const* d_in, const int* in_sizes, int n_in,
                              void* d_out, int out_size, void* d_ws, size_t ws_size,
                              hipStream_t stream) {
  (void)in_sizes; (void)n_in; (void)out_size; (void)ws_size;
  const float* q     = (const float*)d_in[0];
  const float* k     = (const float*)d_in[1];
  const float* v     = (const float*)d_in[2];
  const int*   idx_q = (const int*)d_in[3];
  const int*   idx_k = (const int*)d_in[4];
  float*       out   = (float*)d_out;

  int* ret_cnt = (int*)d_ws;           // H*NB ints
  int* ret_idx = ret_cnt + H_*NB_;     // H*NB*8 ints

  dim3 grid(H_*NB_);
  absa_mask_kernel<<<grid, 256, 0, stream>>>(q, k, idx_q, idx_k, ret_cnt, ret_idx);
  absa_attn_kernel<<<grid, 256, 0, stream>>>(q, k, v, ret_cnt, ret_idx, out);
}